// RGAT_8718783611252
// MI455X (gfx1250) — compile-verified
//
#include <hip/hip_runtime.h>
#include <hip/hip_bf16.h>
#include <stdint.h>

#define D 64
#define NTHREADS 256
#define MAX_REL 32

typedef unsigned int v4u __attribute__((ext_vector_type(4)));
typedef int v4i __attribute__((ext_vector_type(4)));
typedef int v8i __attribute__((ext_vector_type(8)));

#if defined(__has_builtin)
#if __has_builtin(__builtin_amdgcn_tensor_load_to_lds) && \
    __has_builtin(__builtin_amdgcn_s_wait_tensorcnt)
#define HAVE_TDM 1
#endif
#endif
#ifndef HAVE_TDM
#define HAVE_TDM 0
#endif

// ---------------- K1: one lane per edge ------------------------------------
__global__ __launch_bounds__(NTHREADS) void k_edges(
    const float* __restrict__ e_cur, const float* __restrict__ rel,
    const int* __restrict__ head, const int* __restrict__ tail,
    const int* __restrict__ etype, float* __restrict__ sums,
    float* __restrict__ counts, int n_edges, int n_rel, int add_counts) {
  __shared__ float lds_rel[MAX_REL * D];

  int nrel = (n_rel < MAX_REL) ? n_rel : MAX_REL;
  unsigned int lbase = (unsigned int)(uintptr_t)(&lds_rel[0]);

#if HAVE_TDM
  // ---- Tensor Data Mover: 2D tile (dim0=64 f32 x dim1=n_rel) -> LDS ----
  if (threadIdx.x < 32) {
    unsigned long long ga = (unsigned long long)(uintptr_t)rel;
    // D# group 0: count=1, lds_addr, global_addr[56:0], type=2 (bits 127:126)
    v4u g0 = {1u, lbase, (unsigned int)(ga & 0xffffffffull),
              (unsigned int)((ga >> 32) & 0x01ffffffull) | 0x80000000u};
    // D# group 1: data_size=4B(2), tensor_dim0=64, tensor_dim1=nrel,
    //             tile_dim0=64, tile_dim1=nrel, dim0_stride=64, dim1_stride=64*nrel
    int stride1 = D * nrel;
    v8i g1 = {(int)0x00020000u,          // workgroup_mask=0, data_size=2
              (int)((unsigned)D << 16),  // abar_addr=0 | tensor_dim0[15:0]=64
              (int)((unsigned)nrel << 16), // dim0[31:16]=0 | tensor_dim1[15:0]
              (int)((unsigned)D << 16),  // dim1[31:16]=0 | tile_dim0=64
              nrel,                      // tile_dim1=nrel | tile_dim2=0
              D,                         // tensor_dim0_stride[31:0]=64
              (int)((unsigned)(stride1 & 0xffff) << 16), // str0[47:32]=0|str1[15:0]
              (int)((unsigned)stride1 >> 16)};           // str1[47:16]
    v4i g2 = {0, 0, 0, 0};
    v4i g3 = {0, 0, 0, 0};
#if __clang_major__ >= 23
    v8i g4 = {0, 0, 0, 0, 0, 0, 0, 0};
    __builtin_amdgcn_tensor_load_to_lds(g0, g1, g2, g3, g4, 0);
#else
    __builtin_amdgcn_tensor_load_to_lds(g0, g1, g2, g3, 0);
#endif
    __builtin_amdgcn_s_wait_tensorcnt(0);
  }
#else
  // ---- fallback: async global->LDS copy engine ----
  int nf4 = nrel * (D / 4);
  for (int i = threadIdx.x; i < nf4; i += NTHREADS) {
    unsigned long long ga = (unsigned long long)(uintptr_t)(rel + i * 4);
    unsigned int la = lbase + (unsigned int)(i * 16);
    asm volatile("global_load_async_to_lds_b128 %0, %1, off"
                 :: "v"(la), "v"(ga) : "memory");
  }
  asm volatile("s_wait_asynccnt 0x0" ::: "memory");
#endif
  __syncthreads();

  int e = blockIdx.x * blockDim.x + threadIdx.x;
  if (e >= n_edges) return;

  int h = __builtin_nontemporal_load(&head[e]);
  int t = __builtin_nontemporal_load(&tail[e]);
  int r = __builtin_nontemporal_load(&etype[e]) - 1;
  r = (r < 0) ? 0 : ((r >= nrel) ? nrel - 1 : r);

  const float4* hp = (const float4*)(e_cur + (size_t)h * D);
  const float4* tp = (const float4*)(e_cur + (size_t)t * D);
  const float4* rp = (const float4*)(&lds_rel[r * D]);

  // ---- pass 1: Gram matrix of (h, t, r) ----
  float hh = 0.f, tt = 0.f, rr = 0.f, ht = 0.f, hr = 0.f, tr = 0.f;
#pragma unroll
  for (int c = 0; c < D / 4; ++c) {
    float4 hv = hp[c], tv = tp[c], rv = rp[c];
    hh += hv.x * hv.x + hv.y * hv.y + hv.z * hv.z + hv.w * hv.w;
    tt += tv.x * tv.x + tv.y * tv.y + tv.z * tv.z + tv.w * tv.w;
    rr += rv.x * rv.x + rv.y * rv.y + rv.z * rv.z + rv.w * rv.w;
    ht += hv.x * tv.x + hv.y * tv.y + hv.z * tv.z + hv.w * tv.w;
    hr += hv.x * rv.x + hv.y * rv.y + hv.z * rv.z + hv.w * rv.w;
    tr += tv.x * rv.x + tv.y * rv.y + tv.z * rv.z + tv.w * rv.w;
  }

  // ---- scalar chain: whole hyperbolic pipeline on Gram entries ----
  const float MINN = 1e-15f;
  float nh = fmaxf(sqrtf(hh), MINN);
  float th0 = tanhf(nh);
  float alpha = th0 / nh;                 // hyper_head H = alpha * h
  float H2 = alpha * alpha * hh;          // |H|^2
  float lamH = 2.0f / fmaxf(1.0f - H2, MINN);

  // expmap(tail, H) -> T' = a1 h + a2 t
  float nt = fmaxf(sqrtf(tt), MINN);
  float tht = tanhf(0.5f * lamH * nt);
  float beta = tht / nt;
  float y2t = beta * beta * tt;
  float xyt = alpha * beta * ht;
  float dent = fmaxf(1.0f + 2.0f * xyt + H2 * y2t, MINN);
  float a1 = (1.0f + 2.0f * xyt + y2t) * alpha / dent;
  float a2 = (1.0f - H2) * beta / dent;

  // expmap(rel, H) -> R' = b1 h + b2 r
  float nr = fmaxf(sqrtf(rr), MINN);
  float thr = tanhf(0.5f * lamH * nr);
  float gmma = thr / nr;
  float y2r = gmma * gmma * rr;
  float xyr = alpha * gmma * hr;
  float denr = fmaxf(1.0f + 2.0f * xyr + H2 * y2r, MINN);
  float b1 = (1.0f + 2.0f * xyr + y2r) * alpha / denr;
  float b2 = (1.0f - H2) * gmma / denr;

  // mobius_add(T', R') -> res0 = c1 h + c2 t + c3 r
  float x2 = a1 * a1 * hh + 2.f * a1 * a2 * ht + a2 * a2 * tt;
  float y2 = b1 * b1 * hh + 2.f * b1 * b2 * hr + b2 * b2 * rr;
  float xy = a1 * b1 * hh + a1 * b2 * hr + a2 * b1 * ht + a2 * b2 * tr;
  float den = fmaxf(1.f + 2.f * xy + x2 * y2, MINN);
  float k1 = (1.f + 2.f * xy + y2) / den;
  float k2 = (1.f - x2) / den;
  float c1 = k1 * a1 + k2 * b1;
  float c2 = k1 * a2;
  float c3 = k2 * b2;

  // project
  float r0sq = c1 * c1 * hh + c2 * c2 * tt + c3 * c3 * rr +
               2.f * (c1 * c2 * ht + c1 * c3 * hr + c2 * c3 * tr);
  float n0 = fmaxf(sqrtf(r0sq), MINN);
  const float maxn = 1.0f - 4e-3f;
  if (n0 > maxn) {
    float s = maxn / n0;
    c1 *= s; c2 *= s; c3 *= s;
    n0 = maxn;
  }

  // logmap(res0, H): sub = mobius_add(-H, res0) = s1 h + s2 t + s3 r
  float y2l = n0 * n0;
  float hdr = c1 * hh + c2 * ht + c3 * hr;
  float xyl = -alpha * hdr;
  float denl = fmaxf(1.f + 2.f * xyl + H2 * y2l, MINN);
  float k1l = (1.f + 2.f * xyl + y2l) / denl;
  float k2l = (1.f - H2) / denl;
  float s1 = -k1l * alpha + k2l * c1;
  float s2 = k2l * c2;
  float s3 = k2l * c3;
  float nsub2 = s1 * s1 * hh + s2 * s2 * tt + s3 * s3 * rr +
                2.f * (s1 * s2 * ht + s1 * s3 * hr + s2 * s3 * tr);
  float nsub = fmaxf(sqrtf(nsub2), MINN);
  float xc = fminf(nsub, 1.0f - 1e-7f);
  float ath = 0.5f * __logf((1.f + xc) / (1.f - xc));
  float fac = (2.0f / lamH) * ath / nsub;
  float f1 = fac * s1;
  float f2 = fac * s2;
  float f3 = fac * s3;

  // ricci = 1e-7 * normalize(t + r)
  float rc = 1e-7f / fmaxf(sqrtf(tt + 2.f * tr + rr), 1e-12f);
  f2 += rc;
  f3 += rc;

  // ---- pass 2: out = relu(f1 h + f2 t + f3 r), scatter-add by head ----
  float* srow = sums + (size_t)h * D;
#pragma unroll
  for (int c = 0; c < D / 4; ++c) {
    float4 hv = hp[c], tv = tp[c], rv = rp[c];
    float o0 = fmaxf(f1 * hv.x + f2 * tv.x + f3 * rv.x, 0.0f);
    float o1 = fmaxf(f1 * hv.y + f2 * tv.y + f3 * rv.y, 0.0f);
    float o2 = fmaxf(f1 * hv.z + f2 * tv.z + f3 * rv.z, 0.0f);
    float o3 = fmaxf(f1 * hv.w + f2 * tv.w + f3 * rv.w, 0.0f);
    atomicAdd(&srow[4 * c + 0], o0);
    atomicAdd(&srow[4 * c + 1], o1);
    atomicAdd(&srow[4 * c + 2], o2);
    atomicAdd(&srow[4 * c + 3], o3);
  }
  if (add_counts) atomicAdd(&counts[h], 1.0f);  // in-degree: hop 0 only
}

// ---------------- K0: init e_cur = res = entity_emb ; sums = counts = 0 ----
__global__ __launch_bounds__(NTHREADS) void k_init(
    const float* __restrict__ ent, float* __restrict__ e_cur,
    float* __restrict__ res, float* __restrict__ sums,
    float* __restrict__ counts, int n_elem, int n_ent) {
  int i = blockIdx.x * blockDim.x + threadIdx.x;
  if (i < n_elem) {
    float v = ent[i];
    e_cur[i] = v;
    res[i]   = v;
    sums[i]  = 0.0f;
  }
  if (i < n_ent) counts[i] = 0.0f;
}

__device__ __forceinline__ float wave_reduce_sum(float v) {
#pragma unroll
  for (int m = 16; m >= 1; m >>= 1) v += __shfl_xor(v, m, 32);
  return v;
}

// ---------------- K2: wave per entity row: mean, l2-normalize, residual ----
__global__ __launch_bounds__(NTHREADS) void k_finalize(
    float* __restrict__ e_cur, float* __restrict__ res,
    float* __restrict__ sums, const float* __restrict__ counts, int n_ent) {
  int gid = blockIdx.x * blockDim.x + threadIdx.x;
  int row = gid >> 5;
  int lane = gid & 31;
  if (row >= n_ent) return;

  float inv_c = 1.0f / fmaxf(counts[row], 1.0f);
  float2* sp = (float2*)(sums + (size_t)row * D);
  float2 v = sp[lane];
  v.x *= inv_c;
  v.y *= inv_c;
  float ss = wave_reduce_sum(v.x * v.x + v.y * v.y);
  float inv_n = 1.0f / fmaxf(sqrtf(ss), 1e-12f);
  float2 ev;
  ev.x = v.x * inv_n;
  ev.y = v.y * inv_n;

  float2* ep = (float2*)(e_cur + (size_t)row * D);
  float2* op = (float2*)(res + (size_t)row * D);
  ep[lane] = ev;
  float2 o = op[lane];
  o.x = 0.5f * o.x + ev.x;   // RES_LAMBDA = 0.5
  o.y = 0.5f * o.y + ev.y;
  op[lane] = o;

  sp[lane] = make_float2(0.f, 0.f);  // reset for next hop / next call
}

extern "C" void kernel_launch(void* const* d_in, const int* in_sizes, int n_in,
                              void* d_out, int out_size, void* d_ws,
                              size_t ws_size, hipStream_t stream) {
  const float* ent  = (const float*)d_in[0];
  const float* rel  = (const float*)d_in[1];
  const int*   eidx = (const int*)d_in[2];
  const int*   etyp = (const int*)d_in[3];

  int n_ent   = in_sizes[0] / D;
  int n_rel   = in_sizes[1] / D;
  int n_edges = in_sizes[3];
  const int* head = eidx;              // edge_index[0]
  const int* tail = eidx + n_edges;    // edge_index[1]

  float* res    = (float*)d_out;
  float* e_cur  = (float*)d_ws;
  float* sums   = e_cur + (size_t)n_ent * D;
  float* counts = sums + (size_t)n_ent * D;

  int n_elem = n_ent * D;
  int g_init = (n_elem + NTHREADS - 1) / NTHREADS;
  int g_edge = (n_edges + NTHREADS - 1) / NTHREADS;
  int g_fin  = (n_ent * 32 + NTHREADS - 1) / NTHREADS;

  k_init<<<g_init, NTHREADS, 0, stream>>>(ent, e_cur, res, sums, counts,
                                          n_elem, n_ent);
  for (int hop = 0; hop < 2; ++hop) {
    k_edges<<<g_edge, NTHREADS, 0, stream>>>(e_cur, rel, head, tail, etyp,
                                             sums, counts, n_edges, n_rel,
                                             hop == 0 ? 1 : 0);
    k_finalize<<<g_fin, NTHREADS, 0, stream>>>(e_cur, res, sums, counts, n_ent);
  }
}